// GatedNeuralBeliefPropagationRNNCell_20813411516872
// MI455X (gfx1250) — compile-verified
//
#include <hip/hip_runtime.h>
#include <cstdint>
#include <cstddef>

// Problem dims (fixed by the reference).
#define BB 32
#define CC 512
#define VV 1024
#define PARTS 16         // c-split for the two cross-C reductions (chunk = 32 checks)
#define ATANH_EPS 1e-7f

// ---------------------------------------------------------------------------
// CDNA5 async global->LDS helpers (gfx1250: ASYNCcnt-tracked DMA into LDS).
// Inline asm is portable across ROCm-7.2 clang-22 and amdgpu-toolchain
// clang-23 (the clang builtins differ in availability/arity between the two).
// LDS byte offsets are relative to the wave's LDS base; the kernel uses only
// dynamic shared memory, so the array starts at LDS offset 0.
// ---------------------------------------------------------------------------
__device__ __forceinline__ void async_ld_b128(const float* gsrc, unsigned lds_byte_off) {
#if defined(__gfx1250__)
    asm volatile("global_load_async_to_lds_b128 %0, %1, off"
                 :: "v"(lds_byte_off), "v"(gsrc)
                 : "memory");
#else
    (void)gsrc; (void)lds_byte_off;
#endif
}

__device__ __forceinline__ void wait_async0() {
#if defined(__gfx1250__)
    asm volatile("s_wait_asynccnt 0" ::: "memory");
#endif
}

// ---------------------------------------------------------------------------
// Pass 2 (listed first so the disasm snippet shows the async-LDS path):
// one wave32 per (b,c) row of V=1024.
//   excl = T[b,v] - states*w_sum*H
//   t    = H ? tanh(excl/2) : 1
//   p    = exclusive product over v (prefix * suffix scans, zero-safe)
//   new_states = 2*atanh(clip(p))
// states/H/w_sum rows staged via async global->LDS (24 x b128, one wait).
// Lane owns v = 32*i + lane  => conflict-free LDS reads, coalesced stores.
// ---------------------------------------------------------------------------
__global__ __launch_bounds__(32) void bp_row_kernel(
    const float* __restrict__ states, const float* __restrict__ H,
    const float* __restrict__ w_sum, const float* __restrict__ T,
    float* __restrict__ new_states) {
    extern __shared__ float smem[];       // [0,VV): states  [VV,2VV): H  [2VV,3VV): w_sum
    const int c = blockIdx.x;
    const int b = blockIdx.y;
    const int lane = threadIdx.x;         // 0..31 (wave32)

    const float* srow = states + ((size_t)b * CC + (size_t)c) * VV;
    const float* hrow = H      + (size_t)c * VV;
    const float* wrow = w_sum  + (size_t)c * VV;

#pragma unroll
    for (int i = 0; i < 8; ++i) {
        const unsigned e = (unsigned)(i * 32 + lane) * 4u;   // float index (x4 per lane)
        async_ld_b128(srow + e, e * 4u);
        async_ld_b128(hrow + e, 4096u + e * 4u);
        async_ld_b128(wrow + e, 8192u + e * 4u);
    }
    wait_async0();   // single-wave workgroup: no barrier needed

    const float* trow = T + (size_t)b * VV;

    // Gated tanh of the extrinsic sums.
    float t[32];
#pragma unroll
    for (int i = 0; i < 32; ++i) {
        const int v  = i * 32 + lane;
        const float h  = smem[VV + v];
        const float ex = trow[v] - smem[v] * smem[2 * VV + v] * h;
        const float ec = fminf(fmaxf(ex, -40.0f), 40.0f);     // tanh saturates; avoid exp overflow
        const float z  = __expf(ec);                          // tanh(x/2) = (e^x - 1)/(e^x + 1)
        const float th = __fdividef(z - 1.0f, z + 1.0f);
        t[i] = (h != 0.0f) ? th : 1.0f;
    }

    // Left-exclusive products P[i] (prefix over v = 32*i + lane, chunk-major order).
    float P[32];
    float run = 1.0f;
#pragma unroll
    for (int i = 0; i < 32; ++i) {
        float inc = t[i];
#pragma unroll
        for (int d = 1; d < 32; d <<= 1) {
            const float y = __shfl_up(inc, d, 32);
            if (lane >= d) inc *= y;
        }
        float ex = __shfl_up(inc, 1, 32);
        if (lane == 0) ex = 1.0f;
        P[i] = run * ex;
        run *= __shfl(inc, 31, 32);       // chunk total
    }

    // Right-exclusive products, combine, atanh, store.
    float runR = 1.0f;
#pragma unroll
    for (int i = 31; i >= 0; --i) {
        float inc = t[i];
#pragma unroll
        for (int d = 1; d < 32; d <<= 1) {
            const float y = __shfl_down(inc, d, 32);
            if (lane < 32 - d) inc *= y;
        }
        float ex = __shfl_down(inc, 1, 32);
        if (lane == 31) ex = 1.0f;

        float p = P[i] * ex * runR;
        runR *= __shfl(inc, 0, 32);       // chunk total (inclusive-from-right at lane 0)

        p = fminf(fmaxf(p, -1.0f + ATANH_EPS), 1.0f - ATANH_EPS);
        const float ns = __logf(__fdividef(1.0f + p, 1.0f - p));   // 2*atanh(p)

        new_states[((size_t)b * CC + c) * VV + (size_t)(i * 32 + lane)] = ns;
    }
}

// ---------------------------------------------------------------------------
// Pass 1a: partial T[p,b,v] = sum_{c in chunk p} states[b,c,v]*w_sum[c,v]*H[c,v]
// float4 (b128) streams: 512 B per wave per load instruction -> ~6 KB in
// flight per wave at unroll 4, enough latency tolerance for 23.3 TB/s with
// 4096 waves. Thread owns 4 consecutive v.
// ---------------------------------------------------------------------------
__global__ __launch_bounds__(256) void vsum_partial_kernel(
    const float* __restrict__ states, const float* __restrict__ H,
    const float* __restrict__ w_sum, float* __restrict__ part) {
    const int v4 = threadIdx.x;               // 0..255 -> floats [4*v4, 4*v4+4)
    const int b  = blockIdx.y;
    const int p  = blockIdx.z;
    const int c0 = p * (CC / PARTS);
    const float4* sp = (const float4*)(states + ((size_t)b * CC + (size_t)c0) * VV) + v4;
    const float4* hp = (const float4*)(H     + (size_t)c0 * VV) + v4;
    const float4* wp = (const float4*)(w_sum + (size_t)c0 * VV) + v4;
    float4 acc = make_float4(0.f, 0.f, 0.f, 0.f);
#pragma unroll 4
    for (int c = 0; c < CC / PARTS; ++c) {
        const float4 s = sp[(size_t)c * (VV / 4)];
        const float4 h = hp[(size_t)c * (VV / 4)];
        const float4 w = wp[(size_t)c * (VV / 4)];
        acc.x += s.x * w.x * h.x;
        acc.y += s.y * w.y * h.y;
        acc.z += s.z * w.z * h.z;
        acc.w += s.w * w.w * h.w;
    }
    ((float4*)(part + ((size_t)p * BB + b) * VV))[v4] = acc;
}

// Pass 1b: T[b,v] = llr[b,v] + sum_p part[p,b,v]  (fixed-order tree => deterministic)
__global__ __launch_bounds__(256) void vsum_finish_kernel(
    const float* __restrict__ llr, const float* __restrict__ part,
    float* __restrict__ T) {
    const int v4 = threadIdx.x;
    const int b  = blockIdx.y;
    float4 acc = ((const float4*)(llr + (size_t)b * VV))[v4];
#pragma unroll
    for (int p = 0; p < PARTS; ++p) {
        const float4 q = ((const float4*)(part + ((size_t)p * BB + b) * VV))[v4];
        acc.x += q.x; acc.y += q.y; acc.z += q.z; acc.w += q.w;
    }
    ((float4*)(T + (size_t)b * VV))[v4] = acc;
}

// ---------------------------------------------------------------------------
// Pass 3a: partial out[p,b,v] = sum_{c in chunk p} new_states*w_out*H
// (new_states re-read is L2-hot: 64 MB written by pass 2, global L2 = 192 MB)
// ---------------------------------------------------------------------------
__global__ __launch_bounds__(256) void out_partial_kernel(
    const float* __restrict__ ns, const float* __restrict__ H,
    const float* __restrict__ w_out, float* __restrict__ part) {
    const int v4 = threadIdx.x;
    const int b  = blockIdx.y;
    const int p  = blockIdx.z;
    const int c0 = p * (CC / PARTS);
    const float4* np = (const float4*)(ns    + ((size_t)b * CC + (size_t)c0) * VV) + v4;
    const float4* hp = (const float4*)(H     + (size_t)c0 * VV) + v4;
    const float4* wp = (const float4*)(w_out + (size_t)c0 * VV) + v4;
    float4 acc = make_float4(0.f, 0.f, 0.f, 0.f);
#pragma unroll 4
    for (int c = 0; c < CC / PARTS; ++c) {
        const float4 s = np[(size_t)c * (VV / 4)];
        const float4 h = hp[(size_t)c * (VV / 4)];
        const float4 w = wp[(size_t)c * (VV / 4)];
        acc.x += s.x * w.x * h.x;
        acc.y += s.y * w.y * h.y;
        acc.z += s.z * w.z * h.z;
        acc.w += s.w * w.w * h.w;
    }
    ((float4*)(part + ((size_t)p * BB + b) * VV))[v4] = acc;
}

// Pass 3b: out[b,v] = llr[b,v] + sum_p part[p,b,v]
__global__ __launch_bounds__(256) void out_finish_kernel(
    const float* __restrict__ llr, const float* __restrict__ part,
    float* __restrict__ out) {
    const int v4 = threadIdx.x;
    const int b  = blockIdx.y;
    float4 acc = ((const float4*)(llr + (size_t)b * VV))[v4];
#pragma unroll
    for (int p = 0; p < PARTS; ++p) {
        const float4 q = ((const float4*)(part + ((size_t)p * BB + b) * VV))[v4];
        acc.x += q.x; acc.y += q.y; acc.z += q.z; acc.w += q.w;
    }
    ((float4*)(out + (size_t)b * VV))[v4] = acc;
}

// ---------------------------------------------------------------------------
extern "C" void kernel_launch(void* const* d_in, const int* in_sizes, int n_in,
                              void* d_out, int out_size, void* d_ws, size_t ws_size,
                              hipStream_t stream) {
    (void)in_sizes; (void)n_in; (void)out_size; (void)ws_size;
    const float* llr    = (const float*)d_in[0];   // [B,V]
    const float* states = (const float*)d_in[1];   // [B,C*V]
    const float* H      = (const float*)d_in[2];   // [C,V]
    const float* w_sum  = (const float*)d_in[3];   // [C*V]
    const float* w_out  = (const float*)d_in[4];   // [C*V]

    float* out = (float*)d_out;                    // [B,1,V]   (32768 floats)
    float* ns  = out + (size_t)BB * VV;            // [B,C*V]   (16.7M floats)

    float* T  = (float*)d_ws;                      // [B,V]
    float* p1 = T  + (size_t)BB * VV;              // [PARTS,B,V]  (2 MB)
    float* p3 = p1 + (size_t)PARTS * BB * VV;      // [PARTS,B,V]  (2 MB)

    const dim3 gpart(1, BB, PARTS);                // 512 blocks, 4096 waves
    const dim3 gfin(1, BB);

    vsum_partial_kernel<<<gpart, 256, 0, stream>>>(states, H, w_sum, p1);
    vsum_finish_kernel<<<gfin, 256, 0, stream>>>(llr, p1, T);
    bp_row_kernel<<<dim3(CC, BB), 32, 3 * VV * sizeof(float), stream>>>(
        states, H, w_sum, T, ns);
    out_partial_kernel<<<gpart, 256, 0, stream>>>(ns, H, w_out, p3);
    out_finish_kernel<<<gfin, 256, 0, stream>>>(llr, p3, out);
}